// RGCN_3839700763193
// MI455X (gfx1250) — compile-verified
//
#include <hip/hip_runtime.h>
#include <hip/hip_bf16.h>

typedef float v2f __attribute__((ext_vector_type(2)));
typedef float v8f __attribute__((ext_vector_type(8)));

#define F_DIM 128   // aggregation feature width (IN == HID == 128)

// ---------------------------------------------------------------------------
// Degree histograms: deg over relate_src (ND), relate_dst (NS), sim_src (ND),
// sim_dst (ND).  Graph is static across layers -> computed once.
// ---------------------------------------------------------------------------
__global__ __launch_bounds__(256) void rgcn_deg_kernel(
    const int* __restrict__ rs, const int* __restrict__ rd,
    const int* __restrict__ ss, const int* __restrict__ sd,
    float* __restrict__ dA, float* __restrict__ dB,
    float* __restrict__ dC, float* __restrict__ dD, int E) {
  int e = blockIdx.x * blockDim.x + threadIdx.x;
  if (e >= E) return;
  atomicAdd(dA + rs[e], 1.0f);
  atomicAdd(dB + rd[e], 1.0f);
  atomicAdd(dC + ss[e], 1.0f);
  atomicAdd(dD + sd[e], 1.0f);
}

__global__ __launch_bounds__(256) void rgcn_rsqrt_kernel(float* __restrict__ d, int n) {
  int i = blockIdx.x * blockDim.x + threadIdx.x;
  if (i >= n) return;
  d[i] = rsqrtf(fmaxf(d[i], 1.0f));   // pad rows have deg=0 -> scale 1.0
}

// ---------------------------------------------------------------------------
// Edge scatter: one wave (32 lanes) per edge; each lane moves a float4 so the
// wave covers the full 128-float row.  Gather is a global_load_b128 from the
// (L2-resident) source table; accumulation uses global_atomic_add_f32.
// agg[dst] += x[src] * rsqrt(deg_out[src])
// ---------------------------------------------------------------------------
__global__ __launch_bounds__(256) void rgcn_scatter_kernel(
    const float* __restrict__ x, const float* __restrict__ rout,
    const int* __restrict__ src, const int* __restrict__ dst,
    float* __restrict__ agg, int E) {
  int e    = (int)((blockIdx.x * 256u + threadIdx.x) >> 5);
  int lane = threadIdx.x & 31;
  if (e >= E) return;
  int s = src[e], d = dst[e];
  float sc = rout[s];
  const float4 v = *reinterpret_cast<const float4*>(x + (size_t)s * F_DIM + lane * 4);
  float* base = agg + (size_t)d * F_DIM + lane * 4;
  atomicAdd(base + 0, v.x * sc);
  atomicAdd(base + 1, v.y * sc);
  atomicAdd(base + 2, v.z * sc);
  atomicAdd(base + 3, v.w * sc);
}

// ---------------------------------------------------------------------------
// Fused (row-scale . A) @ W + b (+ prev) (+ relu) via V_WMMA_F32_16X16X4_F32.
// One wave computes one 16x16 output tile.  A-fragment layout (ISA 7.12.2):
// lane<16 holds row m0+lane, K={k,k+1}; lane>=16 holds same rows, K={k+2,k+3}.
// B mirrors this with N striped across lanes.  Two accumulators break the
// back-to-back D->C dependency chain.  Tile early-out is wave-uniform so EXEC
// is all-ones through the WMMA region; only the stores are predicated.
// ---------------------------------------------------------------------------
__global__ __launch_bounds__(256) void rgcn_gemm_wmma_kernel(
    const float* __restrict__ A,    // [Mpad x 128], pad rows zeroed
    const float* __restrict__ rin,  // [Mpad] row scale (rsqrt deg_in)
    const float* __restrict__ W,    // [128 x N]
    const float* __restrict__ bias, // [N]
    float* __restrict__ Out,        // [M x N]
    int M, int Mtiles, int N, int accumulate, int relu) {
  const int K   = F_DIM;
  const int nNt = N >> 4;
  int wave = blockIdx.x * 8 + (threadIdx.x >> 5);
  if (wave >= Mtiles * nNt) return;                 // wave-uniform
  int tm = wave / nNt, tn = wave - tm * nNt;
  int lane = threadIdx.x & 31;
  int lo = lane & 15, hi = lane >> 4;
  int m0 = tm << 4, n0 = tn << 4;

  const float* Arow = A + (size_t)(m0 + lo) * K;
  const float* Bcol = W + (n0 + lo);
  float rsc = rin[m0 + lo];

  v8f acc0 = {0.f, 0.f, 0.f, 0.f, 0.f, 0.f, 0.f, 0.f};
  v8f acc1 = {0.f, 0.f, 0.f, 0.f, 0.f, 0.f, 0.f, 0.f};
#pragma unroll 4
  for (int k0 = 0; k0 < K; k0 += 8) {
    int ka = k0 + 2 * hi;
    v2f a, b;
    a.x = Arow[ka] * rsc;            a.y = Arow[ka + 1] * rsc;
    b.x = Bcol[(size_t)ka * N];      b.y = Bcol[(size_t)(ka + 1) * N];
    acc0 = __builtin_amdgcn_wmma_f32_16x16x4_f32(
        false, a, false, b, (short)0, acc0, false, false);
    int kb = ka + 4;
    a.x = Arow[kb] * rsc;            a.y = Arow[kb + 1] * rsc;
    b.x = Bcol[(size_t)kb * N];      b.y = Bcol[(size_t)(kb + 1) * N];
    acc1 = __builtin_amdgcn_wmma_f32_16x16x4_f32(
        false, a, false, b, (short)0, acc1, false, false);
  }
  v8f acc = acc0 + acc1;

  float bn = bias[n0 + lo];
#pragma unroll
  for (int r = 0; r < 8; ++r) {
    int m = m0 + r + 8 * hi;         // C/D layout: vgpr r -> row r (+8 for hi)
    if (m < M) {
      size_t off = (size_t)m * N + (n0 + lo);
      float v = acc[r] + bn;
      if (accumulate) v += Out[off];
      if (relu) v = fmaxf(v, 0.0f);
      Out[off] = v;
    }
  }
}

// ---------------------------------------------------------------------------
extern "C" void kernel_launch(void* const* d_in, const int* in_sizes, int n_in,
                              void* d_out, int out_size, void* d_ws, size_t ws_size,
                              hipStream_t stream) {
  const float* embed_drug = (const float*)d_in[0];
  const float* embed_side = (const float*)d_in[1];
  const int*   rs = (const int*)d_in[2];   // relate_src (drug)
  const int*   rd = (const int*)d_in[3];   // relate_dst (side)
  const int*   ss = (const int*)d_in[4];   // sim_src (drug)
  const int*   sd = (const int*)d_in[5];   // sim_dst (drug)
  const float* W1 = (const float*)d_in[6];
  const float* b1 = (const float*)d_in[7];
  const float* W2 = (const float*)d_in[8];
  const float* b2 = (const float*)d_in[9];
  const float* W3 = (const float*)d_in[10];
  const float* b3 = (const float*)d_in[11];

  const int IN = 128, OUT = 64;
  const int ND = in_sizes[0] / IN;
  const int NS = in_sizes[1] / IN;
  const int E  = in_sizes[2];
  const int NDp = (ND + 15) & ~15;
  const int NSp = (NS + 15) & ~15;

  // ---- carve workspace (256B aligned chunks) ----
  char* wp = (char*)d_ws;
  auto carve = [&](size_t bytes) -> float* {
    float* p = (float*)wp;
    wp += (bytes + 255) & ~(size_t)255;
    return p;
  };
  float* rA    = carve((size_t)NDp * 4);            // rsqrt deg(relate_src)
  float* rB    = carve((size_t)NSp * 4);            // rsqrt deg(relate_dst)
  float* rC    = carve((size_t)NDp * 4);            // rsqrt deg(sim_src)
  float* rD    = carve((size_t)NDp * 4);            // rsqrt deg(sim_dst)
  float* agg_d = carve((size_t)NDp * F_DIM * 4);
  float* agg_s = carve((size_t)NSp * F_DIM * 4);
  float* hd1   = carve((size_t)NDp * F_DIM * 4);
  float* hs1   = carve((size_t)NSp * F_DIM * 4);
  float* hd2   = carve((size_t)NDp * F_DIM * 4);
  float* hs2   = carve((size_t)NSp * F_DIM * 4);

  // ---- degrees (once; graph static across layers) ----
  hipMemsetAsync(rA, 0, (size_t)NDp * 4, stream);
  hipMemsetAsync(rB, 0, (size_t)NSp * 4, stream);
  hipMemsetAsync(rC, 0, (size_t)NDp * 4, stream);
  hipMemsetAsync(rD, 0, (size_t)NDp * 4, stream);
  rgcn_deg_kernel<<<(E + 255) / 256, 256, 0, stream>>>(rs, rd, ss, sd, rA, rB, rC, rD, E);
  rgcn_rsqrt_kernel<<<(NDp + 255) / 256, 256, 0, stream>>>(rA, NDp);
  rgcn_rsqrt_kernel<<<(NSp + 255) / 256, 256, 0, stream>>>(rB, NSp);
  rgcn_rsqrt_kernel<<<(NDp + 255) / 256, 256, 0, stream>>>(rC, NDp);
  rgcn_rsqrt_kernel<<<(NDp + 255) / 256, 256, 0, stream>>>(rD, NDp);

  const int scatter_blocks = (E + 7) / 8;  // one wave per edge, 8 waves/block
  auto gemm_blocks = [](int Mtiles, int N) { return (Mtiles * (N >> 4) + 7) / 8; };

  // One hetero layer: W is (4, 128, N), b is (4, N).
  auto layer = [&](const float* xd, const float* xs, const float* Wt, const float* bt,
                   int N, float* outd, float* outs, int relu) {
    // conv0: relate drug->side
    hipMemsetAsync(agg_s, 0, (size_t)NSp * F_DIM * 4, stream);
    rgcn_scatter_kernel<<<scatter_blocks, 256, 0, stream>>>(xd, rA, rs, rd, agg_s, E);
    rgcn_gemm_wmma_kernel<<<gemm_blocks(NSp / 16, N), 256, 0, stream>>>(
        agg_s, rB, Wt + 0 * F_DIM * N, bt + 0 * N, outs, NS, NSp / 16, N, 0, relu);
    // conv1: relate-by side->drug (src=relate_dst, dst=relate_src)
    hipMemsetAsync(agg_d, 0, (size_t)NDp * F_DIM * 4, stream);
    rgcn_scatter_kernel<<<scatter_blocks, 256, 0, stream>>>(xs, rB, rd, rs, agg_d, E);
    rgcn_gemm_wmma_kernel<<<gemm_blocks(NDp / 16, N), 256, 0, stream>>>(
        agg_d, rA, Wt + 1 * F_DIM * N, bt + 1 * N, outd, ND, NDp / 16, N, 0, 0);
    // conv2: similar drug->drug
    hipMemsetAsync(agg_d, 0, (size_t)NDp * F_DIM * 4, stream);
    rgcn_scatter_kernel<<<scatter_blocks, 256, 0, stream>>>(xd, rC, ss, sd, agg_d, E);
    rgcn_gemm_wmma_kernel<<<gemm_blocks(NDp / 16, N), 256, 0, stream>>>(
        agg_d, rD, Wt + 2 * F_DIM * N, bt + 2 * N, outd, ND, NDp / 16, N, 1, 0);
    // conv3: similar-by (src=sim_dst, dst=sim_src)
    hipMemsetAsync(agg_d, 0, (size_t)NDp * F_DIM * 4, stream);
    rgcn_scatter_kernel<<<scatter_blocks, 256, 0, stream>>>(xd, rD, sd, ss, agg_d, E);
    rgcn_gemm_wmma_kernel<<<gemm_blocks(NDp / 16, N), 256, 0, stream>>>(
        agg_d, rC, Wt + 3 * F_DIM * N, bt + 3 * N, outd, ND, NDp / 16, N, 1, relu);
  };

  float* out_hd = (float*)d_out;                       // [ND x 64]
  float* out_hs = (float*)d_out + (size_t)ND * OUT;    // [NS x 64]

  layer(embed_drug, embed_side, W1, b1, 128, hd1, hs1, /*relu=*/1);
  layer(hd1, hs1, W2, b2, 128, hd2, hs2, /*relu=*/1);
  layer(hd2, hs2, W3, b3, 64, out_hd, out_hs, /*relu=*/0);
}